// GatedMetaFusion_59931973649030
// MI455X (gfx1250) — compile-verified
//
#include <hip/hip_runtime.h>

// ---------------------------------------------------------------------------
// GatedMetaFusion on gfx1250 (MI455X): fp32 WMMA (16x16x4) row-MLPs.
// ---------------------------------------------------------------------------

typedef __attribute__((ext_vector_type(2))) float v2f;
typedef __attribute__((ext_vector_type(8))) float v8f;

#define DIMF   128
#define ASTR   34       // LDS K-stride (dwords); 34 -> conflict-free frag reads
#define KPANEL 32       // K elements staged per panel (8 wmma k-steps)
#define ROWS   128      // rows (nodes/edges) per workgroup supertile
#define NTILES 8        // 128 output cols / 16

static __device__ __forceinline__ v8f wmma4(v2f a, v2f b, v8f c) {
  // D = A(16x4) x B(4x16) + C(16x16), fp32
  return __builtin_amdgcn_wmma_f32_16x16x4_f32(false, a, false, b, (short)0, c,
                                               false, false);
}

// Stage transposed weight panel: Bs[col*ASTR + kk] = W[(kbase+kk)*128 + col]
static __device__ __forceinline__ void stage_w(float* Bs,
                                               const float* __restrict__ W,
                                               int kbase, int Klim, int tid) {
  for (int e = tid; e < DIMF * KPANEL; e += 256) {
    int kk = e >> 7;          // 0..31
    int col = e & 127;        // 0..127 (coalesced global read)
    int k = kbase + kk;
    Bs[col * ASTR + kk] = (k < Klim) ? W[k * DIMF + col] : 0.0f;
  }
}

// One K-panel of WMMAs: wave 'wv' owns rows [16*wv,16*wv+16), all 8 col-tiles.
static __device__ __forceinline__ void mma_panel(const float* As,
                                                 const float* Bs, int wv,
                                                 int lane, v8f* acc) {
  const int m = lane & 15;
  const int sel = (lane & 16) ? 2 : 0;
  const float* ap = As + (wv * 16 + m) * ASTR + sel;  // A: lane = row M
  const float* bp = Bs + m * ASTR + sel;              // B: lane = col N
#pragma unroll
  for (int kk = 0; kk < KPANEL; kk += 4) {
    v2f a = *(const v2f*)(ap + kk);                   // ds_load_b64
#pragma unroll
    for (int j = 0; j < NTILES; ++j) {
      v2f b = *(const v2f*)(bp + j * 16 * ASTR + kk); // ds_load_b64
      acc[j] = wmma4(a, b, acc[j]);
    }
  }
}

static __device__ __forceinline__ void init_bias(v8f* acc,
                                                 const float* __restrict__ b,
                                                 int lane) {
#pragma unroll
  for (int j = 0; j < NTILES; ++j) {
    float bv = b[j * 16 + (lane & 15)];
    v8f c;
#pragma unroll
    for (int r = 0; r < 8; ++r) c[r] = bv;
    acc[j] = c;
  }
}

static __device__ __forceinline__ void relu_acc(v8f* acc) {
#pragma unroll
  for (int j = 0; j < NTILES; ++j)
#pragma unroll
    for (int r = 0; r < 8; ++r) acc[j][r] = fmaxf(acc[j][r], 0.0f);
}

// Write hidden activations (C layout) for cols [32p,32p+32) back to As.
static __device__ __forceinline__ void store_h_panel(float* As, const v8f* h,
                                                     int p, int wv, int lane) {
  const int mB = (lane & 16) ? 8 : 0;
  const int n = lane & 15;
#pragma unroll
  for (int jj = 0; jj < 2; ++jj) {
    int j = p * 2 + jj;
#pragma unroll
    for (int r = 0; r < 8; ++r)
      As[(wv * 16 + mB + r) * ASTR + jj * 16 + n] = h[j][r];
  }
}

// ---------------------------------------------------------------------------
// Kernel 1: per-edge gate2 MLP + scatter-accumulate into sums/cnt
// ---------------------------------------------------------------------------
__global__ __launch_bounds__(256) void k_gate2(
    const float* __restrict__ res, const float* __restrict__ meta,
    const int* __restrict__ edge, const float* __restrict__ vec,
    const float* __restrict__ W1, const float* __restrict__ b1,
    const float* __restrict__ W2, const float* __restrict__ b2,
    float* __restrict__ sums, float* __restrict__ cnt, int E) {
  __shared__ float As[ROWS * ASTR];
  __shared__ float Bs[DIMF * ASTR];
  __shared__ int srcI[ROWS];
  __shared__ int dstI[ROWS];

  const int tid = threadIdx.x;
  const int lane = tid & 31;
  const int wv = tid >> 5;
  const int base = blockIdx.x * ROWS;

  if (tid < ROWS) {
    int gr = base + tid;
    int s = 0, d = 0;
    if (gr < E) { s = edge[gr]; d = edge[E + gr]; }
    srcI[tid] = s;
    dstI[tid] = d;
  }
  __syncthreads();

  v8f acc1[NTILES];
  init_bias(acc1, b1, lane);

  // layer 1: K = 259 (2*128+3), padded to 9 panels of 32
  for (int p = 0; p < 9; ++p) {
    int kbase = p * KPANEL;
    __syncthreads();
    for (int e = tid; e < ROWS * KPANEL; e += 256) {
      int r = e >> 5, kk = e & 31;
      int k = kbase + kk;
      int gr = base + r;
      float v = 0.0f;
      if (gr < E) {
        if (k < 128)      v = res[dstI[r] * DIMF + k];
        else if (k < 256) v = meta[srcI[r] * DIMF + (k - 128)];
        else if (k < 259) v = vec[gr * 3 + (k - 256)];
      }
      As[r * ASTR + kk] = v;
    }
    stage_w(Bs, W1, kbase, 259, tid);
    __syncthreads();
    mma_panel(As, Bs, wv, lane, acc1);
  }
  relu_acc(acc1);

  // layer 2: K = 128, 4 panels; reuse As (h) and Bs (W2^T)
  v8f acc2[NTILES];
  init_bias(acc2, b2, lane);
  for (int p = 0; p < 4; ++p) {
    __syncthreads();
    store_h_panel(As, acc1, p, wv, lane);
    stage_w(Bs, W2, p * KPANEL, 128, tid);
    __syncthreads();
    mma_panel(As, Bs, wv, lane, acc2);
  }

  // epilogue: vals = gate2_w * meta[src]; atomic scatter into sums / cnt
  const int mB = (lane & 16) ? 8 : 0;
  const int n0 = lane & 15;
#pragma unroll
  for (int j = 0; j < NTILES; ++j) {
#pragma unroll
    for (int r = 0; r < 8; ++r) {
      int lrow = wv * 16 + mB + r;
      int gr = base + lrow;
      if (gr < E) {
        int n = j * 16 + n0;
        float mv = meta[srcI[lrow] * DIMF + n];
        atomicAdd(&sums[dstI[lrow] * DIMF + n], acc2[j][r] * mv);
      }
    }
  }
  if (lane < 16) {
    int lrow = wv * 16 + lane;
    int gr = base + lrow;
    if (gr < E) atomicAdd(&cnt[dstI[lrow]], 1.0f);
  }
}

// ---------------------------------------------------------------------------
// Kernel 2: per-node gate1 MLP, then fused = res + g1w*meta_b + sums/max(cnt,1)
// ---------------------------------------------------------------------------
__global__ __launch_bounds__(256) void k_gate1_fuse(
    const float* __restrict__ res, const float* __restrict__ meta,
    const int* __restrict__ sec, const float* __restrict__ pe,
    const float* __restrict__ W1, const float* __restrict__ b1,
    const float* __restrict__ W2, const float* __restrict__ b2,
    const float* __restrict__ sums, const float* __restrict__ cnt,
    float* __restrict__ out, int N) {
  __shared__ float As[ROWS * ASTR];
  __shared__ float Bs[DIMF * ASTR];
  __shared__ int secI[ROWS];

  const int tid = threadIdx.x;
  const int lane = tid & 31;
  const int wv = tid >> 5;
  const int base = blockIdx.x * ROWS;

  if (tid < ROWS) {
    int gr = base + tid;
    secI[tid] = (gr < N) ? sec[gr] : 0;
  }
  __syncthreads();

  v8f acc1[NTILES];
  init_bias(acc1, b1, lane);
  for (int p = 0; p < 9; ++p) {
    int kbase = p * KPANEL;
    __syncthreads();
    for (int e = tid; e < ROWS * KPANEL; e += 256) {
      int r = e >> 5, kk = e & 31;
      int k = kbase + kk;
      int gr = base + r;
      float v = 0.0f;
      if (gr < N) {
        if (k < 128)      v = res[gr * DIMF + k];
        else if (k < 256) v = meta[secI[r] * DIMF + (k - 128)];
        else if (k < 259) v = -pe[gr * 3 + (k - 256)];
      }
      As[r * ASTR + kk] = v;
    }
    stage_w(Bs, W1, kbase, 259, tid);
    __syncthreads();
    mma_panel(As, Bs, wv, lane, acc1);
  }
  relu_acc(acc1);

  v8f acc2[NTILES];
  init_bias(acc2, b2, lane);
  for (int p = 0; p < 4; ++p) {
    __syncthreads();
    store_h_panel(As, acc1, p, wv, lane);
    stage_w(Bs, W2, p * KPANEL, 128, tid);
    __syncthreads();
    mma_panel(As, Bs, wv, lane, acc2);
  }

  const int mB = (lane & 16) ? 8 : 0;
  const int n0 = lane & 15;
#pragma unroll
  for (int j = 0; j < NTILES; ++j) {
#pragma unroll
    for (int r = 0; r < 8; ++r) {
      int lrow = wv * 16 + mB + r;
      int gr = base + lrow;
      if (gr < N) {
        int n = j * 16 + n0;
        float mb = meta[secI[lrow] * DIMF + n];
        float s = sums[gr * DIMF + n];
        float c = cnt[gr];
        out[gr * DIMF + n] =
            res[gr * DIMF + n] + acc2[j][r] * mb + s / fmaxf(c, 1.0f);
      }
    }
  }
}

// ---------------------------------------------------------------------------
// Kernel 3: final fuse MLP (128->128->128), in place on d_out
// ---------------------------------------------------------------------------
__global__ __launch_bounds__(256) void k_fusemlp(
    const float* __restrict__ W1, const float* __restrict__ b1,
    const float* __restrict__ W2, const float* __restrict__ b2,
    float* __restrict__ io, int N) {
  __shared__ float As[ROWS * ASTR];
  __shared__ float Bs[DIMF * ASTR];

  const int tid = threadIdx.x;
  const int lane = tid & 31;
  const int wv = tid >> 5;
  const int base = blockIdx.x * ROWS;

  v8f acc1[NTILES];
  init_bias(acc1, b1, lane);
  for (int p = 0; p < 4; ++p) {
    int kbase = p * KPANEL;
    __syncthreads();
    for (int e = tid; e < ROWS * KPANEL; e += 256) {
      int r = e >> 5, kk = e & 31;
      int gr = base + r;
      As[r * ASTR + kk] = (gr < N) ? io[gr * DIMF + kbase + kk] : 0.0f;
    }
    stage_w(Bs, W1, kbase, 128, tid);
    __syncthreads();
    mma_panel(As, Bs, wv, lane, acc1);
  }
  relu_acc(acc1);

  v8f acc2[NTILES];
  init_bias(acc2, b2, lane);
  for (int p = 0; p < 4; ++p) {
    __syncthreads();
    store_h_panel(As, acc1, p, wv, lane);
    stage_w(Bs, W2, p * KPANEL, 128, tid);
    __syncthreads();
    mma_panel(As, Bs, wv, lane, acc2);
  }

  const int mB = (lane & 16) ? 8 : 0;
  const int n0 = lane & 15;
#pragma unroll
  for (int j = 0; j < NTILES; ++j) {
#pragma unroll
    for (int r = 0; r < 8; ++r) {
      int gr = base + wv * 16 + mB + r;
      if (gr < N) io[gr * DIMF + j * 16 + n0] = acc2[j][r];
    }
  }
}

__global__ void k_zero(float* __restrict__ p, long n) {
  long i = (long)blockIdx.x * blockDim.x + threadIdx.x;
  long stride = (long)gridDim.x * blockDim.x;
  for (; i < n; i += stride) p[i] = 0.0f;
}

// ---------------------------------------------------------------------------
extern "C" void kernel_launch(void* const* d_in, const int* in_sizes, int n_in,
                              void* d_out, int out_size, void* d_ws,
                              size_t ws_size, hipStream_t stream) {
  const float* res  = (const float*)d_in[0];
  const float* meta = (const float*)d_in[1];
  const int*   sec  = (const int*)d_in[2];
  const float* pe   = (const float*)d_in[3];
  const int*   edge = (const int*)d_in[4];
  const float* vec  = (const float*)d_in[5];
  const float* g1W1 = (const float*)d_in[6];
  const float* g1b1 = (const float*)d_in[7];
  const float* g1W2 = (const float*)d_in[8];
  const float* g1b2 = (const float*)d_in[9];
  const float* g2W1 = (const float*)d_in[10];
  const float* g2b1 = (const float*)d_in[11];
  const float* g2W2 = (const float*)d_in[12];
  const float* g2b2 = (const float*)d_in[13];
  const float* fW1  = (const float*)d_in[14];
  const float* fb1  = (const float*)d_in[15];
  const float* fW2  = (const float*)d_in[16];
  const float* fb2  = (const float*)d_in[17];

  const int N = in_sizes[0] / DIMF;
  const int E = in_sizes[4] / 2;

  float* sums = (float*)d_ws;          // N*128 floats
  float* cnt  = sums + (long)N * DIMF; // N floats
  float* out  = (float*)d_out;

  k_zero<<<2048, 256, 0, stream>>>(sums, (long)N * DIMF + N);
  k_gate2<<<(E + ROWS - 1) / ROWS, 256, 0, stream>>>(
      res, meta, edge, vec, g2W1, g2b1, g2W2, g2b2, sums, cnt, E);
  k_gate1_fuse<<<(N + ROWS - 1) / ROWS, 256, 0, stream>>>(
      res, meta, sec, pe, g1W1, g1b1, g1W2, g1b2, sums, cnt, out, N);
  k_fusemlp<<<(N + ROWS - 1) / ROWS, 256, 0, stream>>>(fW1, fb1, fW2, fb2, out,
                                                       N);
}